// EBM_12189117186447
// MI455X (gfx1250) — compile-verified
//
#include <hip/hip_runtime.h>
#include <math.h>

// ---------------------------------------------------------------------------
// EBM contrastive loss for MI455X (gfx1250, wave32).
//  K1: txt row-normalize -> tn                       (2 MB stream, b64 loads)
//  K2: vid per-frame normalize + frame mean -> vmean (25 MB stream, b64 loads)
//  K3: E = -(tn @ vmean^T) via V_WMMA_F32_16X16X4_F32, fused tile reduction
//  K4: 30-step Langevin MCMC, register-resident, shuffle-butterfly reductions
//  K5: deterministic tree reduction + loss assembly -> d_out[0]
// ---------------------------------------------------------------------------

typedef __attribute__((ext_vector_type(2))) float v2f;
typedef __attribute__((ext_vector_type(8))) float v8f;

#define Bsz   1024
#define Fr    12
#define Dm    512
#define TPB   256
#define NWAVE (TPB / 32)
#define TILES 64            // Bsz/16
#define MCMC_STEPS 30
#define STEP_SIZE  0.01f
#define NOISE_AMP  0.005f
#define COEF_REG   0.001f
#define EPSN       1e-12f

// ---------------- deterministic counter-based Gaussian ---------------------
__device__ __forceinline__ unsigned pcg_hash(unsigned x) {
    unsigned state = x * 747796405u + 2891336453u;
    unsigned word  = ((state >> ((state >> 28u) + 4u)) ^ state) * 277803737u;
    return (word >> 22u) ^ word;
}
__device__ __forceinline__ void gauss2(unsigned seed, float* z0, float* z1) {
    unsigned h1 = pcg_hash(seed);
    unsigned h2 = pcg_hash(h1 ^ 0x9E3779B9u);
    float u1 = (float)(h1 + 1u) * 2.3283064e-10f;   // (0,1]
    float u2 = (float)h2        * 2.3283064e-10f;
    float r  = sqrtf(-2.0f * logf(u1));
    float a  = 6.28318530718f * u2;
    *z0 = r * cosf(a);
    *z1 = r * sinf(a);
}

// -------- two-value block reduction: wave butterfly + 8-slot LDS stage -----
// On return every thread holds the block-wide sums. 2 barriers total.
__device__ __forceinline__ void block_reduce2(float& a, float& b,
                                              float* __restrict__ xw /*2*NWAVE*/) {
#pragma unroll
    for (int off = 16; off > 0; off >>= 1) {
        a += __shfl_xor(a, off, 32);
        b += __shfl_xor(b, off, 32);
    }
    const int wave = threadIdx.x >> 5;
    const int lane = threadIdx.x & 31;
    if (lane == 0) { xw[wave * 2] = a; xw[wave * 2 + 1] = b; }
    __syncthreads();
    float ra = 0.f, rb = 0.f;
#pragma unroll
    for (int w = 0; w < NWAVE; ++w) { ra += xw[w * 2]; rb += xw[w * 2 + 1]; }
    __syncthreads();                       // xw reusable after this
    a = ra; b = rb;
}

// ---------------- K1: normalize txt rows -----------------------------------
__global__ void txt_norm_kernel(const float* __restrict__ txt,
                                float* __restrict__ tn) {
    __shared__ float xw[2 * NWAVE];
    const int row = blockIdx.x, tid = threadIdx.x;
    const float2* x2 = (const float2*)(txt + row * Dm);
    float2 a = x2[tid];
    float ss = a.x * a.x + a.y * a.y, dummy = 0.f;
    block_reduce2(ss, dummy, xw);
    float inv = 1.0f / fmaxf(sqrtf(ss), EPSN);
    float2* o2 = (float2*)(tn + row * Dm);
    o2[tid] = make_float2(a.x * inv, a.y * inv);
}

// ---------------- K2: per-frame normalize + mean over frames ----------------
__global__ void vmean_kernel(const float* __restrict__ vid,
                             float* __restrict__ vmean) {
    __shared__ float xw[2 * NWAVE];
    const int b = blockIdx.x, tid = threadIdx.x;
    float acc0 = 0.0f, acc1 = 0.0f;
#pragma unroll 1
    for (int f = 0; f < Fr; ++f) {
        const float2* x2 = (const float2*)(vid + (b * Fr + f) * Dm);
        float2 a = x2[tid];
        float ss = a.x * a.x + a.y * a.y, dummy = 0.f;
        block_reduce2(ss, dummy, xw);
        float inv = 1.0f / fmaxf(sqrtf(ss), EPSN);
        acc0 += a.x * inv;
        acc1 += a.y * inv;
    }
    const float cF = 1.0f / (float)Fr;
    float2* o2 = (float2*)(vmean + b * Dm);
    o2[tid] = make_float2(acc0 * cF, acc1 * cF);
}

// ---------------- K3: WMMA GEMM E = -(tn @ vmean^T) + tile stats ------------
// grid (64,16), block 128 = 4 waves; wave w -> tile (blockIdx.x, blockIdx.y*4+w)
// tilebuf[tile*4 + {0,1,2,3}] = {sumE, sumE2, diagE, diagE2} for that tile.
__global__ void energy_gemm_kernel(const float* __restrict__ tn,
                                   const float* __restrict__ vmean,
                                   float* __restrict__ tilebuf) {
    const int lane  = threadIdx.x & 31;
    const int wave  = threadIdx.x >> 5;
    const int tileM = blockIdx.x;                  // a index (txt rows)
    const int tileN = blockIdx.y * 4 + wave;       // b index (vid cols)
    const int half  = lane >> 4;
    const int l16   = lane & 15;

    // A (16x4, f32): lane<16 holds M=l16 K={0,1}; lane>=16 K={2,3}
    // B (4x16, f32): mirrored layout, row index = N column of the tile
    const float* ap = tn    + (tileM * 16 + l16) * Dm + 2 * half;
    const float* bp = vmean + (tileN * 16 + l16) * Dm + 2 * half;

    v8f c = {};
#pragma unroll 8
    for (int k = 0; k < Dm; k += 4) {
        v2f av, bv;
        av.x = ap[k];  av.y = ap[k + 1];
        bv.x = bp[k];  bv.y = bp[k + 1];
        c = __builtin_amdgcn_wmma_f32_16x16x4_f32(
                false, av, false, bv, (short)0, c, false, false);
    }

    // C/D layout: reg r, lane<16 -> (M=r, N=lane); lane>=16 -> (M=r+8, N=lane-16)
    float s = 0.f, s2 = 0.f, ds = 0.f, ds2 = 0.f;
    const bool diag_tile = (tileM == tileN);
#pragma unroll
    for (int r = 0; r < 8; ++r) {
        float e = -c[r];
        s  += e;
        s2 += e * e;
        int m = r + 8 * half;
        if (diag_tile && m == l16) { ds += e; ds2 += e * e; }
    }
#pragma unroll
    for (int off = 16; off > 0; off >>= 1) {
        s   += __shfl_xor(s,   off, 32);
        s2  += __shfl_xor(s2,  off, 32);
        ds  += __shfl_xor(ds,  off, 32);
        ds2 += __shfl_xor(ds2, off, 32);
    }
    if (lane == 0) {
        float* o = tilebuf + (tileM * TILES + tileN) * 4;
        o[0] = s; o[1] = s2; o[2] = ds; o[3] = ds2;
    }
}

// ---------------- K4: resident Langevin MCMC, 1 block per sample ------------
__global__ void mcmc_kernel(const float* __restrict__ vid_init,
                            const float* __restrict__ txt_init,
                            float* __restrict__ esbuf) {
    __shared__ float xw[2 * NWAVE];
    const int a = blockIdx.x, tid = threadIdx.x;
    const float cF = 1.0f / (float)Fr;

    const float2* t2 = (const float2*)(txt_init + a * Dm);
    float2 ti = t2[tid];
    float t0 = 2.f * ti.x - 1.f;
    float t1 = 2.f * ti.y - 1.f;
    float v0[Fr], v1[Fr];
#pragma unroll
    for (int f = 0; f < Fr; ++f) {
        const float2* v2p = (const float2*)(vid_init + (a * Fr + f) * Dm);
        float2 vi = v2p[tid];
        v0[f] = 2.f * vi.x - 1.f;
        v1[f] = 2.f * vi.y - 1.f;
    }

#pragma unroll 1
    for (int step = 0; step < MCMC_STEPS; ++step) {
        float tss = t0 * t0 + t1 * t1, dummy = 0.f;
        block_reduce2(tss, dummy, xw);
        float it = 1.0f / fmaxf(sqrtf(tss), EPSN);

        float gt0 = 0.f, gt1 = 0.f;
#pragma unroll 1
        for (int f = 0; f < Fr; ++f) {
            float vv = v0[f] * v0[f] + v1[f] * v1[f];   // ||v_f||^2
            float vt = v0[f] * t0    + v1[f] * t1;      // v_f . t
            block_reduce2(vv, vt, xw);
            float iv = 1.0f / fmaxf(sqrtf(vv), EPSN);
            float sf = vt * iv * it;                    // cos(v_f, t)

            // grads at the current point
            float gv0 = -cF * (t0 * it - sf * v0[f] * iv) * iv;
            float gv1 = -cF * (t1 * it - sf * v1[f] * iv) * iv;
            gt0 += -cF * (v0[f] * iv - sf * t0 * it) * it;
            gt1 += -cF * (v1[f] * iv - sf * t1 * it) * it;

            float z0, z1;
            gauss2((((unsigned)step * Bsz + a) * (Fr + 1) + f) * TPB + tid, &z0, &z1);
            v0[f] = v0[f] - STEP_SIZE * gv0 + NOISE_AMP * z0;
            v1[f] = v1[f] - STEP_SIZE * gv1 + NOISE_AMP * z1;
        }
        float zt0, zt1;
        gauss2((((unsigned)step * Bsz + a) * (Fr + 1) + Fr) * TPB + tid + 0x5bd1e995u,
               &zt0, &zt1);
        t0 = t0 - STEP_SIZE * gt0 + NOISE_AMP * zt0;
        t1 = t1 - STEP_SIZE * gt1 + NOISE_AMP * zt1;
    }

    // e_s[a] = -(1/F) * sum_f cos(v_f, t) at final state
    float tss = t0 * t0 + t1 * t1, dummy = 0.f;
    block_reduce2(tss, dummy, xw);
    float it = 1.0f / fmaxf(sqrtf(tss), EPSN);

    float esum = 0.f;
#pragma unroll 1
    for (int f = 0; f < Fr; ++f) {
        float vv = v0[f] * v0[f] + v1[f] * v1[f];
        float vt = v0[f] * t0    + v1[f] * t1;
        block_reduce2(vv, vt, xw);
        float iv = 1.0f / fmaxf(sqrtf(vv), EPSN);
        esum += vt * iv * it;
    }
    if (tid == 0) {
        float e = -esum * cF;
        esbuf[a * 2 + 0] = e;
        esbuf[a * 2 + 1] = e * e;
    }
}

// ---------------- K5: deterministic reduction + loss ------------------------
__global__ void reduce_finalize_kernel(const float* __restrict__ tilebuf,
                                       const float* __restrict__ esbuf,
                                       float* __restrict__ out) {
    __shared__ float red[6 * TPB];
    const int tid = threadIdx.x;
    float s = 0.f, s2 = 0.f, ds = 0.f, ds2 = 0.f, es = 0.f, es2 = 0.f;
    for (int i = tid; i < TILES * TILES; i += TPB) {
        s   += tilebuf[i * 4 + 0];
        s2  += tilebuf[i * 4 + 1];
        ds  += tilebuf[i * 4 + 2];
        ds2 += tilebuf[i * 4 + 3];
    }
    for (int i = tid; i < Bsz; i += TPB) {
        es  += esbuf[i * 2 + 0];
        es2 += esbuf[i * 2 + 1];
    }
    red[0 * TPB + tid] = s;   red[1 * TPB + tid] = s2;
    red[2 * TPB + tid] = ds;  red[3 * TPB + tid] = ds2;
    red[4 * TPB + tid] = es;  red[5 * TPB + tid] = es2;
    __syncthreads();
    for (int st = TPB / 2; st > 0; st >>= 1) {
        if (tid < st)
            for (int k = 0; k < 6; ++k)
                red[k * TPB + tid] += red[k * TPB + tid + st];
        __syncthreads();
    }
    if (tid == 0) {
        float sumE = red[0], sumE2 = red[TPB], sumD = red[2 * TPB],
              sumD2 = red[3 * TPB], sumEs = red[4 * TPB], sumEs2 = red[5 * TPB];
        const float Bf = (float)Bsz;
        const float n_neg = Bf * (Bf - 1.0f);
        float pos_mean    = sumD  / Bf;
        float pos_sq_mean = sumD2 / Bf;
        float neg_mean    = (sumE  - sumD)  / n_neg;
        float neg_sq_mean = (sumE2 - sumD2) / n_neg;
        float loss1 = pos_mean - neg_mean + COEF_REG * (pos_sq_mean + neg_sq_mean);
        float es_mean    = sumEs  / Bf;
        float es_sq_mean = sumEs2 / Bf;
        float loss2 = pos_mean - es_mean + COEF_REG * (pos_sq_mean + es_sq_mean);
        out[0] = 0.5f * (loss1 + loss2);
    }
}

// ---------------------------------------------------------------------------
extern "C" void kernel_launch(void* const* d_in, const int* in_sizes, int n_in,
                              void* d_out, int out_size, void* d_ws, size_t ws_size,
                              hipStream_t stream) {
    const float* vid      = (const float*)d_in[0];   // [1024,12,512]
    const float* txt      = (const float*)d_in[1];   // [1024,512]
    const float* vid_init = (const float*)d_in[2];   // [1024,12,512]
    const float* txt_init = (const float*)d_in[3];   // [1024,512]
    float* out = (float*)d_out;

    // workspace layout (floats): tn | vmean | tilebuf | esbuf  (~4.3 MB)
    float* ws      = (float*)d_ws;
    float* tn      = ws;
    float* vmean   = tn + (size_t)Bsz * Dm;
    float* tilebuf = vmean + (size_t)Bsz * Dm;
    float* esbuf   = tilebuf + (size_t)TILES * TILES * 4;

    txt_norm_kernel<<<Bsz, TPB, 0, stream>>>(txt, tn);
    vmean_kernel<<<Bsz, TPB, 0, stream>>>(vid, vmean);
    energy_gemm_kernel<<<dim3(TILES, TILES / 4), 128, 0, stream>>>(tn, vmean, tilebuf);
    mcmc_kernel<<<Bsz, TPB, 0, stream>>>(vid_init, txt_init, esbuf);
    reduce_finalize_kernel<<<1, TPB, 0, stream>>>(tilebuf, esbuf, out);
}